// Model_24575802867956
// MI455X (gfx1250) — compile-verified
//
#include <hip/hip_runtime.h>

typedef __attribute__((ext_vector_type(2))) float v2f;
typedef __attribute__((ext_vector_type(8))) float v8f;

#define F 128
#define WPB 8        // waves per block (256 threads, wave32)
#define SCAN_T 256
#define SCAN_E 4
#define SCAN_B (SCAN_T * SCAN_E)   // 1024 elements per scan block

// ---------------- utility: zero int buffer ----------------
__global__ void k_zero_i32(int* __restrict__ p, int n) {
  int i = blockIdx.x * blockDim.x + threadIdx.x;
  int stride = gridDim.x * blockDim.x;
  for (; i < n; i += stride) p[i] = 0;
}

// ---------------- CSR build: degree count ----------------
__global__ void k_count(const int* __restrict__ dst, int* __restrict__ deg, int E) {
  int e = blockIdx.x * blockDim.x + threadIdx.x;
  if (e < E) atomicAdd(&deg[dst[e]], 1);
}

// ---------------- CSR build: two-level exclusive scan of deg -> offs ----------------
__global__ void k_scan1(const int* __restrict__ deg, int* __restrict__ offs,
                        int* __restrict__ bsums, int N) {
  __shared__ int sh[SCAN_T];
  int base = blockIdx.x * SCAN_B + threadIdx.x * SCAN_E;
  int v[SCAN_E];
  int lsum = 0;
  #pragma unroll
  for (int j = 0; j < SCAN_E; ++j) {
    int idx = base + j;
    int d = (idx < N) ? deg[idx] : 0;
    v[j] = lsum;            // exclusive prefix within thread
    lsum += d;
  }
  sh[threadIdx.x] = lsum;
  __syncthreads();
  for (int off = 1; off < SCAN_T; off <<= 1) {
    int t = (threadIdx.x >= off) ? sh[threadIdx.x - off] : 0;
    __syncthreads();
    sh[threadIdx.x] += t;
    __syncthreads();
  }
  int texcl = (threadIdx.x > 0) ? sh[threadIdx.x - 1] : 0;
  #pragma unroll
  for (int j = 0; j < SCAN_E; ++j) {
    int idx = base + j;
    if (idx < N) offs[idx] = texcl + v[j];
  }
  if (threadIdx.x == SCAN_T - 1) bsums[blockIdx.x] = sh[SCAN_T - 1];
}

// single block; requires nb <= SCAN_T (N=100000 -> nb=98)
__global__ void k_scan2(int* __restrict__ bsums, int nb) {
  __shared__ int sh[SCAN_T];
  int v = (threadIdx.x < nb) ? bsums[threadIdx.x] : 0;
  sh[threadIdx.x] = v;
  __syncthreads();
  for (int off = 1; off < SCAN_T; off <<= 1) {
    int t = (threadIdx.x >= off) ? sh[threadIdx.x - off] : 0;
    __syncthreads();
    sh[threadIdx.x] += t;
    __syncthreads();
  }
  if (threadIdx.x < nb) bsums[threadIdx.x] = (threadIdx.x > 0) ? sh[threadIdx.x - 1] : 0;
}

__global__ void k_scan3(int* __restrict__ offs, const int* __restrict__ bsums, int N) {
  int i = blockIdx.x * blockDim.x + threadIdx.x;
  if (i < N) offs[i] += bsums[i / SCAN_B];
}

// ---------------- CSR build: fill incident-source lists ----------------
__global__ void k_fill(const int* __restrict__ src, const int* __restrict__ dst,
                       const int* __restrict__ offs, int* __restrict__ cursor,
                       int* __restrict__ eidx, int E) {
  int e = blockIdx.x * blockDim.x + threadIdx.x;
  if (e < E) {
    int d = dst[e];
    int pos = offs[d] + atomicAdd(&cursor[d], 1);
    eidx[pos] = src[e];
  }
}

// ---------------- gather-based mean aggregation ----------------
// one wave per node, float4 per lane; mean (1/max(deg,1)) fused
__global__ void __launch_bounds__(256)
k_aggregate(const float* __restrict__ h, const int* __restrict__ eidx,
            const int* __restrict__ offs, const int* __restrict__ deg,
            float* __restrict__ hneigh, int N) {
  int n = blockIdx.x * WPB + (threadIdx.x >> 5);
  if (n >= N) return;
  int lane = threadIdx.x & 31;
  int beg = offs[n];
  int d = deg[n];
  float4 acc = {0.f, 0.f, 0.f, 0.f};
  for (int j = 0; j < d; ++j) {
    int s = eidx[beg + j];   // uniform across wave -> scalar load
    const float4 hv = ((const float4*)(h + (long long)s * F))[lane];
    acc.x += hv.x; acc.y += hv.y; acc.z += hv.z; acc.w += hv.w;
  }
  float sc = 1.0f / fmaxf((float)d, 1.0f);
  float4 o = {acc.x * sc, acc.y * sc, acc.z * sc, acc.w * sc};
  ((float4*)(hneigh + (long long)n * F))[lane] = o;
}

// ---------------- fused SAGE layer GEMM (fp32 WMMA) ----------------
// hout = hin @ Wself + hneigh @ Wneigh + bias
// A 16x4 f32 frag: lane m = lane&15, K-offset = 2*(lane>>4)
// B 4x16 f32 frag: lane n = lane&15, K = vgpr + 2*(lane>>4)
// C/D 16x16 f32:  M = v + 8*(lane>>4), N = lane&15
__global__ void __launch_bounds__(256)
k_sage_gemm(const float* __restrict__ hin, const float* __restrict__ hneigh,
            const float* __restrict__ Wself, const float* __restrict__ Wneigh,
            const float* __restrict__ bias, float* __restrict__ hout) {
  const int lane = threadIdx.x & 31;
  const int wave = threadIdx.x >> 5;
  const int row0 = blockIdx.x * 16;
  const int col0 = wave * 16;
  const int m    = lane & 15;
  const int koff = (lane >> 4) << 1;   // 0 or 2

  v8f c;
  {
    float bv = bias[col0 + m];         // bias depends only on output column N
    #pragma unroll
    for (int v = 0; v < 8; ++v) c[v] = bv;
  }

  // pass 1: A = hin, B = Wself
  const float* Ar = hin + (long long)(row0 + m) * F;
  #pragma unroll 4
  for (int k0 = 0; k0 < F; k0 += 4) {
    int k = k0 + koff;
    v2f a; a.x = Ar[k];                   a.y = Ar[k + 1];
    v2f b; b.x = Wself[k * F + col0 + m]; b.y = Wself[(k + 1) * F + col0 + m];
    c = __builtin_amdgcn_wmma_f32_16x16x4_f32(false, a, false, b, (short)0, c,
                                              false, false);
  }

  // pass 2: A = hneigh (already mean-scaled), B = Wneigh
  const float* Mr = hneigh + (long long)(row0 + m) * F;
  #pragma unroll 4
  for (int k0 = 0; k0 < F; k0 += 4) {
    int k = k0 + koff;
    v2f a; a.x = Mr[k];                    a.y = Mr[k + 1];
    v2f b; b.x = Wneigh[k * F + col0 + m]; b.y = Wneigh[(k + 1) * F + col0 + m];
    c = __builtin_amdgcn_wmma_f32_16x16x4_f32(false, a, false, b, (short)0, c,
                                              false, false);
  }

  const int mbase = row0 + ((lane >> 4) << 3);
  #pragma unroll
  for (int v = 0; v < 8; ++v)
    hout[(long long)(mbase + v) * F + col0 + m] = c[v];
}

// ---------------- edge dot: s[e] = dot(h[src[e]], h[dst[e]]) ----------------
__global__ void __launch_bounds__(256)
k_edge_dot(const float* __restrict__ h, const int* __restrict__ src,
           const int* __restrict__ dst, float* __restrict__ s, int E) {
  int e = blockIdx.x * WPB + (threadIdx.x >> 5);
  if (e >= E) return;
  int lane = threadIdx.x & 31;
  const float4 a = ((const float4*)(h + (long long)src[e] * F))[lane];
  const float4 b = ((const float4*)(h + (long long)dst[e] * F))[lane];
  float p = a.x * b.x + a.y * b.y + a.z * b.z + a.w * b.w;
  #pragma unroll
  for (int off = 16; off > 0; off >>= 1) p += __shfl_xor(p, off, 32);
  if (lane == 0) s[e] = p;
}

// ---------------- min/max over scores (monotone uint encoding) ----------------
__device__ __forceinline__ unsigned f2ord(float f) {
  unsigned u = __float_as_uint(f);
  return (u & 0x80000000u) ? ~u : (u | 0x80000000u);
}
__device__ __forceinline__ float ord2f(unsigned k) {
  return __uint_as_float((k & 0x80000000u) ? (k & 0x7FFFFFFFu) : ~k);
}

__global__ void k_minmax_init(unsigned* mm) {
  if (threadIdx.x == 0) { mm[0] = 0xFFFFFFFFu; mm[1] = 0u; }
}

__global__ void __launch_bounds__(256)
k_minmax(const float* __restrict__ s, int E, unsigned* __restrict__ mm) {
  __shared__ unsigned smn[256], smx[256];
  unsigned mn = 0xFFFFFFFFu, mx = 0u;
  for (int i = blockIdx.x * blockDim.x + threadIdx.x; i < E;
       i += gridDim.x * blockDim.x) {
    unsigned k = f2ord(s[i]);
    mn = min(mn, k); mx = max(mx, k);
  }
  smn[threadIdx.x] = mn; smx[threadIdx.x] = mx;
  __syncthreads();
  for (int off = 128; off > 0; off >>= 1) {
    if (threadIdx.x < off) {
      smn[threadIdx.x] = min(smn[threadIdx.x], smn[threadIdx.x + off]);
      smx[threadIdx.x] = max(smx[threadIdx.x], smx[threadIdx.x + off]);
    }
    __syncthreads();
  }
  if (threadIdx.x == 0) { atomicMin(&mm[0], smn[0]); atomicMax(&mm[1], smx[0]); }
}

__global__ void k_normalize(float* __restrict__ s, int E,
                            const unsigned* __restrict__ mm) {
  int i = blockIdx.x * blockDim.x + threadIdx.x;
  if (i >= E) return;
  float mn = ord2f(mm[0]);
  float mx = ord2f(mm[1]);
  float inv = 1.0f / (mx - mn);
  s[i] = (s[i] - mn) * inv;
}

// ---------------- driver ----------------
extern "C" void kernel_launch(void* const* d_in, const int* in_sizes, int n_in,
                              void* d_out, int out_size, void* d_ws, size_t ws_size,
                              hipStream_t stream) {
  const float* x   = (const float*)d_in[0];
  const int*   src = (const int*)d_in[1];
  const int*   dst = (const int*)d_in[2];
  const float* Ws1 = (const float*)d_in[3];
  const float* Wn1 = (const float*)d_in[4];
  const float* b1  = (const float*)d_in[5];
  const float* Ws2 = (const float*)d_in[6];
  const float* Wn2 = (const float*)d_in[7];
  const float* b2  = (const float*)d_in[8];
  float* out = (float*)d_out;

  const int N = in_sizes[0] / F;   // 100000 (multiple of 16)
  const int E = in_sizes[1];       // 1600000

  const long long nf = (long long)N * F;
  // workspace layout
  float* hneigh = (float*)d_ws;
  float* h1     = hneigh + nf;
  float* h2     = h1 + nf;
  int*   deg    = (int*)(h2 + nf);
  int*   offs   = deg + N;
  int*   cursor = offs + N;
  int*   eidx   = cursor + N;
  int*   bsums  = eidx + E;
  unsigned* mm  = (unsigned*)(bsums + SCAN_T);

  const int eblocks = (E + 255) / 256;
  const int wblocks = (E + WPB - 1) / WPB;       // wave-per-edge kernels
  const int nwaves  = (N + WPB - 1) / WPB;       // wave-per-node kernels
  const int gblocks = N / 16;                    // GEMM row stripes
  const int sblocks = (N + SCAN_B - 1) / SCAN_B; // scan level-1 blocks (98 <= 256)

  // ---- build CSR once (shared by both layers) ----
  k_zero_i32<<<512, 256, 0, stream>>>(deg, 2 * N);          // deg + offs (adjacent)
  k_count<<<eblocks, 256, 0, stream>>>(dst, deg, E);
  k_scan1<<<sblocks, SCAN_T, 0, stream>>>(deg, offs, bsums, N);
  k_scan2<<<1, SCAN_T, 0, stream>>>(bsums, sblocks);
  k_scan3<<<(N + 255) / 256, 256, 0, stream>>>(offs, bsums, N);
  k_fill<<<eblocks, 256, 0, stream>>>(src, dst, offs, cursor, eidx, E);

  // ---- layer 1 ----
  k_aggregate<<<nwaves, 256, 0, stream>>>(x, eidx, offs, deg, hneigh, N);
  k_sage_gemm<<<gblocks, 256, 0, stream>>>(x, hneigh, Ws1, Wn1, b1, h1);

  // ---- layer 2 ----
  k_aggregate<<<nwaves, 256, 0, stream>>>(h1, eidx, offs, deg, hneigh, N);
  k_sage_gemm<<<gblocks, 256, 0, stream>>>(h1, hneigh, Ws2, Wn2, b2, h2);

  // ---- edge scores + minmax normalize (in place in d_out) ----
  k_edge_dot<<<wblocks, 256, 0, stream>>>(h2, src, dst, out, E);
  k_minmax_init<<<1, 32, 0, stream>>>(mm);
  k_minmax<<<1024, 256, 0, stream>>>(out, E, mm);
  k_normalize<<<eblocks, 256, 0, stream>>>(out, E, mm);
}